// PCGraphConv_80736795230658
// MI455X (gfx1250) — compile-verified
//
#include <hip/hip_runtime.h>
#include <cstdint>

// Predictive-coding graph message passing for MI455X (gfx1250, wave32).
// B==32 == wave width: node state kept node-major [N][32] so each edge is a
// contiguous 128B gather + 128B atomic-scatter per wave. All node arrays are
// 512KB -> fully L2 resident (192MB L2); edge metadata is streamed into LDS
// with the CDNA5 Tensor Data Mover (double buffered, s_wait_tensorcnt).

#define NPG   4096
#define EPG   524288
#define BATCHSZ 32
#define BN    (BATCHSZ * NPG)
#define CHUNK 256                 // edges staged per TDM transfer
#define SC_BLOCKS 1024
#define SC_THREADS 256

typedef unsigned int u32x4 __attribute__((ext_vector_type(4)));
typedef int i32x4 __attribute__((ext_vector_type(4)));
typedef int i32x8 __attribute__((ext_vector_type(8)));

// ---- Tensor Data Mover: 1-row tile of `nelem` 4-byte elements -> LDS ------
__device__ __forceinline__ void tdm_load_1d(const void* gptr, void* lds_ptr,
                                            uint32_t nelem) {
  uint64_t ga = (uint64_t)(uintptr_t)gptr;
  uint32_t la = (uint32_t)(uintptr_t)lds_ptr;   // low 32 bits of flat LDS addr == LDS byte offset
  u32x4 g0;
  g0.x = 1u;                                    // count=1, user mode, gather off
  g0.y = la;                                    // lds_addr
  g0.z = (uint32_t)ga;                          // global_addr[31:0]
  g0.w = (uint32_t)((ga >> 32) & 0x01FFFFFFu)   // global_addr[56:32]
         | (2u << 30);                          // type=2 ("image")
  i32x8 g1;
  g1[0] = (2 << 16);                            // wg_mask=0 (no cluster), data_size=2 -> 4B
  g1[1] = (int)((nelem & 0xFFFFu) << 16);       // tensor_dim0[15:0]
  g1[2] = (int)(((nelem >> 16) & 0xFFFFu)       // tensor_dim0[31:16]
         | (1u << 16));                         // tensor_dim1 = 1 (keep dim-1 index 0 in bounds)
  g1[3] = (int)((nelem & 0xFFFFu) << 16);       // tensor_dim1 hi=0 | tile_dim0 = nelem
  g1[4] = 0;                                    // tile_dim1 = tile_dim2 = 0 (unused)
  g1[5] = (int)nelem;                           // tensor_dim0_stride (irrelevant, 1 row)
  g1[6] = 0;
  g1[7] = 0;
  i32x4 g2 = {1, 1, 0, 0};                      // tensor_dim2=1, tensor_dim3=1, tile_dim3=0
  i32x4 g3 = {0, 1 << 16, 0, 0};                // tensor_dim4=1, tile_dim4=0
  i32x8 g4 = {0, 0, 0, 0, 0, 0, 0, 0};          // clang-23 extra group (zero-filled)
  __builtin_amdgcn_tensor_load_to_lds(g0, g1, g2, g3, g4, 0);
}

// ---- K1: batch-major x -> node-major tanh(x), x; zero both accumulators ---
__global__ void k_prep(const float* __restrict__ x, float* __restrict__ fx_t,
                       float* __restrict__ x_t, float* __restrict__ mu_t,
                       float* __restrict__ acc2) {
  int tid = blockIdx.x * blockDim.x + threadIdx.x;
  if (tid >= BN) return;
  int b = tid >> 12;              // / NPG
  int n = tid & (NPG - 1);
  float v  = x[tid];              // coalesced read
  float fx = tanhf(v);
  int i = n * BATCHSZ + b;        // node-major
  fx_t[i] = fx;
  x_t[i]  = v;
  mu_t[tid] = 0.0f;               // re-zero accumulators every call (replay safe)
  acc2[tid] = 0.0f;
}

// ---- K2/K4: per-edge gather/scale/atomic-scatter, TDM-staged metadata -----
// For each edge e: gdst[sidx[e]*32 + lane] += w[e] * gsrc[gidx[e]*32 + lane]
__global__ void k_scatter(const int* __restrict__ gidx, const int* __restrict__ sidx,
                          const float* __restrict__ ew,
                          const float* __restrict__ gsrc, float* __restrict__ gdst) {
  __shared__ int lds_s[2 * CHUNK];
  __shared__ int lds_d[2 * CHUNK];
  __shared__ int lds_w[2 * CHUNK];
  const int lane = threadIdx.x & 31;
  const int wid  = threadIdx.x >> 5;            // 0..7
  const int nChunks = EPG / CHUNK;              // 2048

  int c = blockIdx.x;
  if (threadIdx.x < 32 && c < nChunks) {        // wave 0 issues DMA (uniform branch)
    tdm_load_1d(gidx + (size_t)c * CHUNK, &lds_s[0], CHUNK);
    tdm_load_1d(sidx + (size_t)c * CHUNK, &lds_d[0], CHUNK);
    tdm_load_1d(ew   + (size_t)c * CHUNK, &lds_w[0], CHUNK);
  }
  int buf = 0;
  for (; c < nChunks; c += gridDim.x) {
    const int cn = c + (int)gridDim.x;
    if (threadIdx.x < 32) {
      if (cn < nChunks) {                       // prefetch next chunk, then wait for current
        const int nb = buf ^ 1;
        tdm_load_1d(gidx + (size_t)cn * CHUNK, &lds_s[nb * CHUNK], CHUNK);
        tdm_load_1d(sidx + (size_t)cn * CHUNK, &lds_d[nb * CHUNK], CHUNK);
        tdm_load_1d(ew   + (size_t)cn * CHUNK, &lds_w[nb * CHUNK], CHUNK);
        __builtin_amdgcn_s_wait_tensorcnt(3);   // in-order: first 3 (current buf) done
      } else {
        __builtin_amdgcn_s_wait_tensorcnt(0);
      }
    }
    __syncthreads();
    // Each wave owns 32 edges; lane k's triple is broadcast via v_readlane.
    const int base = buf * CHUNK + wid * 32;
    int myS = lds_s[base + lane];
    int myD = lds_d[base + lane];
    int myW = lds_w[base + lane];
#pragma unroll
    for (int k = 0; k < 32; ++k) {
      int   s  = __builtin_amdgcn_readlane(myS, k);   // -> SGPR
      int   d  = __builtin_amdgcn_readlane(myD, k);
      float wv = __int_as_float(__builtin_amdgcn_readlane(myW, k));
      float v  = wv * gsrc[(size_t)s * BATCHSZ + lane];        // 128B coalesced gather
      unsafeAtomicAdd(&gdst[(size_t)d * BATCHSZ + lane], v);   // global_atomic_add_f32
    }
    __syncthreads();                            // all waves done before buf is re-filled
    buf ^= 1;
  }
}

// ---- K3: eps = x - mu (node-major); emit mu in batch-major output ---------
__global__ void k_eps(const float* __restrict__ x_t, const float* __restrict__ mu_t,
                      float* __restrict__ eps_t, float* __restrict__ out_mu) {
  int i = blockIdx.x * blockDim.x + threadIdx.x;
  if (i >= BN) return;
  float mu = mu_t[i];
  eps_t[i] = x_t[i] - mu;
  int n = i >> 5, b = i & 31;
  out_mu[b * NPG + n] = mu;
}

// ---- K5: dx = -eps + (1 - fx^2) * backflow; emit batch-major --------------
__global__ void k_dx(const float* __restrict__ fx_t, const float* __restrict__ eps_t,
                     const float* __restrict__ acc2, float* __restrict__ out_dx) {
  int i = blockIdx.x * blockDim.x + threadIdx.x;
  if (i >= BN) return;
  float fx = fx_t[i];
  float dx = -eps_t[i] + (1.0f - fx * fx) * acc2[i];
  int n = i >> 5, b = i & 31;
  out_dx[b * NPG + n] = dx;
}

extern "C" void kernel_launch(void* const* d_in, const int* in_sizes, int n_in,
                              void* d_out, int out_size, void* d_ws, size_t ws_size,
                              hipStream_t stream) {
  (void)in_sizes; (void)n_in; (void)out_size; (void)ws_size;
  const float* x  = (const float*)d_in[0];
  const float* w  = (const float*)d_in[1];
  const int*   es = (const int*)d_in[2];
  const int*   ed = (const int*)d_in[3];

  float* out_mu = (float*)d_out;
  float* out_dx = out_mu + BN;

  float* fx_t  = (float*)d_ws;          // 5 node-major buffers, 512KB each (2.5MB)
  float* x_t   = fx_t  + BN;
  float* mu_t  = x_t   + BN;
  float* eps_t = mu_t  + BN;
  float* acc2  = eps_t + BN;

  dim3 blk(256);
  k_prep<<<BN / 256, blk, 0, stream>>>(x, fx_t, x_t, mu_t, acc2);
  // mu pass: gather tanh(x) at src, scatter-add at dst
  k_scatter<<<SC_BLOCKS, dim3(SC_THREADS), 0, stream>>>(es, ed, w, fx_t, mu_t);
  k_eps<<<BN / 256, blk, 0, stream>>>(x_t, mu_t, eps_t, out_mu);
  // value pass: gather eps at dst, scatter-add at src
  k_scatter<<<SC_BLOCKS, dim3(SC_THREADS), 0, stream>>>(ed, es, w, eps_t, acc2);
  k_dx<<<BN / 256, blk, 0, stream>>>(fx_t, eps_t, acc2, out_dx);
}